// THGLP_59322088292962
// MI455X (gfx1250) — compile-verified
//
#include <hip/hip_runtime.h>
#include <hip/hip_bf16.h>
#include <math.h>

#define CN   30000
#define CE   30000
#define CNNZ 100000
#define CM   10000
#define CT   8
#define CL   4

typedef __attribute__((ext_vector_type(2))) float v2f;
typedef __attribute__((ext_vector_type(8))) float v8f;

// ---------------------------------------------------------------------------
// fp32 WMMA GEMM: C[rows x ncols] = A[rows x K] * B[K x ncols] (+bias, relu).
// rows % 16 == 0, K % 32 == 0, ncols % (NT*16) == 0 (dispatched by host).
// Block = 8 waves; each wave owns a 16 x (NT*16) strip of C. The B column
// slab (32 K-rows) is staged in LDS as K-PAIRS: bs[p][c] = (B[2p][c],
// B[2p+1][c]) stored as one v2f, so a lane's B fragment is a single aligned
// ds_load_b64 directly into the WMMA operand register pair (no repack movs).
// Pair-rows are skewed by +16 v2f so the two 16-lane halves of the wave hit
// disjoint 32-bank sets (conflict-free b64 loads). A fragments for the slab
// are batch-loaded into registers (global_load_b64). All branches are
// wave-uniform; EXEC stays all-ones for every WMMA.
// ---------------------------------------------------------------------------
template <int NT>
__global__ __launch_bounds__(256) void gemm_f32_wmma(
    const float* __restrict__ A, const float* __restrict__ B, float* __restrict__ C,
    int rows, int K, int lda, int ldb, int ldc,
    const float* __restrict__ bias, int relu)
{
  constexpr int KS    = 32;        // K-rows per LDS slab
  constexpr int NC    = NT * 16;   // columns per block strip
  constexpr int NPAIR = KS / 2;    // K-pairs per slab
  constexpr int PSTR  = NC + 16;   // pair-row stride in v2f (bank de-skew)
  __shared__ v2f bs[NPAIR * PSTR];

  const int wave = threadIdx.x >> 5;
  const int lane = threadIdx.x & 31;
  const int half = lane >> 4;      // K/M half of the fragment this lane holds
  const int lr   = lane & 15;
  const int m0   = blockIdx.x * 128 + wave * 16;
  const int nb   = blockIdx.y * NC;

  v8f acc[NT];
#pragma unroll
  for (int t = 0; t < NT; ++t) acc[t] = (v8f){0.f,0.f,0.f,0.f,0.f,0.f,0.f,0.f};

  // A fragment: lane holds A[m0+lr][kk + 2*half + {0,1}] (8B-aligned -> b64).
  // Clamp row for out-of-range waves (they still run barriers + WMMA).
  const int arowi = (m0 + lr < rows) ? (m0 + lr) : (rows - 1);
  const float* arow = A + (size_t)arowi * lda + 2 * half;

  for (int k0 = 0; k0 < K; k0 += KS) {
    __syncthreads();                       // protect previous slab reads
    // cooperative B slab load (two coalesced rows per K-pair)
#pragma unroll
    for (int e = threadIdx.x; e < NPAIR * NC; e += 256) {
      const int p = e / NC;
      const int c = e - p * NC;
      v2f pr;
      pr.x = B[(size_t)(k0 + 2 * p)     * ldb + nb + c];
      pr.y = B[(size_t)(k0 + 2 * p + 1) * ldb + nb + c];
      bs[p * PSTR + c] = pr;
    }
    // batched A fragment loads for the whole slab
    v2f af[KS / 4];
#pragma unroll
    for (int u = 0; u < KS / 4; ++u)
      af[u] = *(const v2f*)(arow + k0 + u * 4);
    __syncthreads();
    // 8 K-steps against LDS-resident B; lane's pair-row = 2u + half
#pragma unroll
    for (int u = 0; u < KS / 4; ++u) {
      const v2f* bsrow = bs + (2 * u + half) * PSTR + lr;
#pragma unroll
      for (int t = 0; t < NT; ++t) {
        const v2f b = bsrow[t * 16];       // single ds_load_b64
        acc[t] = __builtin_amdgcn_wmma_f32_16x16x4_f32(
            false, af[u], false, b, (short)0, acc[t], false, false);
      }
    }
  }

  if (m0 < rows) {                         // wave-uniform store guard
    const int rbase = m0 + half * 8;       // C layout: VGPR j -> row 8*half + j
#pragma unroll
    for (int t = 0; t < NT; ++t) {
      const int col = nb + t * 16 + lr;
      const float bv = bias ? bias[col] : 0.0f;
#pragma unroll
      for (int j = 0; j < 8; ++j) {
        float v = acc[t][j] + bv;
        if (relu) v = fmaxf(v, 0.0f);
        C[(size_t)(rbase + j) * ldc + col] = v;
      }
    }
  }
}

// ---------------------------------------------------------------------------
// Elementwise / segment helpers
// ---------------------------------------------------------------------------
__global__ void zero_f32(float* __restrict__ p, int n) {
  int i = blockIdx.x * blockDim.x + threadIdx.x;
  if (i < n) p[i] = 0.0f;
}

__global__ void count_deg(const int* __restrict__ idx, float* __restrict__ deg, int n) {
  int i = blockIdx.x * blockDim.x + threadIdx.x;
  if (i < n) atomicAdd(&deg[idx[i]], 1.0f);
}

__global__ void invert_f32(float* __restrict__ p, int n) {
  int i = blockIdx.x * blockDim.x + threadIdx.x;
  if (i < n) { float v = p[i]; p[i] = (v > 0.0f) ? (1.0f / v) : 0.0f; }
}

// dst[sidx[i]*F + f] += src[gidx[i]*F + f] * (scale ? scale[gidx[i]] : 1)
__global__ void scatter_add(const float* __restrict__ src, float* __restrict__ dst,
                            const int* __restrict__ gidx, const int* __restrict__ sidx,
                            const float* __restrict__ scale, int logF, int total) {
  int tid = blockIdx.x * blockDim.x + threadIdx.x;
  if (tid >= total) return;
  const int i = tid >> logF;
  const int f = tid & ((1 << logF) - 1);
  const int s = gidx[i];
  const int d = sidx[i];
  float v = src[((size_t)s << logF) + f];
  if (scale) v *= scale[s];
  atomicAdd(&dst[((size_t)d << logF) + f], v);
}

// x = relu(acc * dinv[row] + b[f])
__global__ void finalize_relu(float* __restrict__ x, const float* __restrict__ acc,
                              const float* __restrict__ dinv, const float* __restrict__ b,
                              int logF, int total) {
  int tid = blockIdx.x * blockDim.x + threadIdx.x;
  if (tid >= total) return;
  const int i = tid >> logF;
  const int f = tid & ((1 << logF) - 1);
  x[tid] = fmaxf(fmaf(acc[tid], dinv[i], b[f]), 0.0f);
}

// dst[m*stride + f] = src[marks[m]*F + f]
__global__ void gather_rows(const float* __restrict__ src, const int* __restrict__ marks,
                            float* __restrict__ dst, int logF, int dstStride, int total) {
  int tid = blockIdx.x * blockDim.x + threadIdx.x;
  if (tid >= total) return;
  const int m = tid >> logF;
  const int f = tid & ((1 << logF) - 1);
  dst[(size_t)m * dstStride + f] = src[((size_t)marks[m] << logF) + f];
}

// a = ex[em], c = ex[em+1]; dmin=min(a,c), dmax=max(a,c)   (F = 32)
__global__ void gather_edge_mm(const float* __restrict__ ex, const int* __restrict__ em,
                               float* __restrict__ dmin, float* __restrict__ dmax,
                               int dstStride, int total) {
  int tid = blockIdx.x * blockDim.x + threadIdx.x;
  if (tid >= total) return;
  const int m = tid >> 5;
  const int f = tid & 31;
  const int e = em[m];
  const float a = ex[(size_t)e * 32 + f];
  const float c = ex[((size_t)e + 1) * 32 + f];
  dmin[(size_t)m * dstStride + f] = fminf(a, c);
  dmax[(size_t)m * dstStride + f] = fmaxf(a, c);
}

// TI[m*4096 + r] += pos[r]   (pos is [T*512] = 4096 contiguous)
__global__ void add_pos(float* __restrict__ TI, const float* __restrict__ pos, int total) {
  int tid = blockIdx.x * blockDim.x + threadIdx.x;
  if (tid < total) TI[tid] += pos[tid & (CT * 512 - 1)];
}

__global__ void transpose512(const float* __restrict__ in, float* __restrict__ out) {
  int tid = blockIdx.x * blockDim.x + threadIdx.x;
  if (tid < 512 * 512) {
    const int r = tid >> 9, c = tid & 511;
    out[(size_t)c * 512 + r] = in[tid];
  }
}

// HIN[m, 0:512] += ff_b + ti[m, 7, :]   (residual)
__global__ void ff_epilogue(float* __restrict__ HIN, const float* __restrict__ TI,
                            const float* __restrict__ ffb, int total) {
  int tid = blockIdx.x * blockDim.x + threadIdx.x;
  if (tid >= total) return;
  const int m = tid >> 9;
  const int f = tid & 511;
  HIN[(size_t)m * 1024 + f] += ffb[f] + TI[(size_t)m * (CT * 512) + 7 * 512 + f];
}

// One wave per m: scores[s] = ti[m,s,:].qk[m,:] / sqrt(T); softmax (row 7 of the
// causal mask is all-ones so no masking); U[m,:] = sum_s attn[s] * ti[m,s,:]
__global__ __launch_bounds__(256) void attn_last_row(
    const float* __restrict__ TI, const float* __restrict__ QK,
    float* __restrict__ U, int M) {
  const int m = blockIdx.x * 8 + (threadIdx.x >> 5);
  const int lane = threadIdx.x & 31;
  if (m >= M) return;
  const float* ti = TI + (size_t)m * (CT * 512);
  const float* qk = QK + (size_t)m * 512;
  float qv[16];
#pragma unroll
  for (int j = 0; j < 16; ++j) qv[j] = qk[lane + 32 * j];
  float sc[CT];
  for (int s = 0; s < CT; ++s) {
    const float* tr = ti + s * 512;
    float p = 0.0f;
#pragma unroll
    for (int j = 0; j < 16; ++j) p += tr[lane + 32 * j] * qv[j];
    p += __shfl_xor(p, 16, 32);
    p += __shfl_xor(p, 8, 32);
    p += __shfl_xor(p, 4, 32);
    p += __shfl_xor(p, 2, 32);
    p += __shfl_xor(p, 1, 32);
    sc[s] = p * 0.35355339059327373f;   // 1/sqrt(8)
  }
  float mx = sc[0];
  for (int s = 1; s < CT; ++s) mx = fmaxf(mx, sc[s]);
  float sum = 0.0f;
  for (int s = 0; s < CT; ++s) { sc[s] = expf(sc[s] - mx); sum += sc[s]; }
  const float inv = 1.0f / sum;
#pragma unroll
  for (int j = 0; j < 16; ++j) {
    float u = 0.0f;
    for (int s = 0; s < CT; ++s) u += sc[s] * ti[s * 512 + lane + 32 * j];
    U[(size_t)m * 512 + lane + 32 * j] = u * inv;
  }
}

// logits = H @ W2 + b2 ; out = log_softmax over 2 classes
__global__ void final_logits(const float* __restrict__ H, const float* __restrict__ W2,
                             const float* __restrict__ b2, float* __restrict__ out, int M) {
  int m = blockIdx.x * blockDim.x + threadIdx.x;
  if (m >= M) return;
  const float* h = H + (size_t)m * 256;
  float z0 = b2[0], z1 = b2[1];
  for (int k = 0; k < 256; ++k) {
    const float hv = h[k];
    z0 = fmaf(hv, W2[2 * k], z0);
    z1 = fmaf(hv, W2[2 * k + 1], z1);
  }
  const float mx = fmaxf(z0, z1);
  const float lse = mx + logf(expf(z0 - mx) + expf(z1 - mx));
  out[2 * m]     = z0 - lse;
  out[2 * m + 1] = z1 - lse;
}

// ---------------------------------------------------------------------------
// Host driver
// ---------------------------------------------------------------------------
static inline void launch_gemm(hipStream_t s, const float* A, const float* B, float* C,
                               int rows, int K, int ncols, int lda, int ldb, int ldc,
                               const float* bias, int relu) {
  if ((ncols & 63) == 0) {
    dim3 grid((rows + 127) / 128, ncols / 64);
    gemm_f32_wmma<4><<<grid, 256, 0, s>>>(A, B, C, rows, K, lda, ldb, ldc, bias, relu);
  } else if ((ncols & 31) == 0) {
    dim3 grid((rows + 127) / 128, ncols / 32);
    gemm_f32_wmma<2><<<grid, 256, 0, s>>>(A, B, C, rows, K, lda, ldb, ldc, bias, relu);
  } else {
    dim3 grid((rows + 127) / 128, ncols / 16);
    gemm_f32_wmma<1><<<grid, 256, 0, s>>>(A, B, C, rows, K, lda, ldb, ldc, bias, relu);
  }
}

extern "C" void kernel_launch(void* const* d_in, const int* in_sizes, int n_in,
                              void* d_out, int out_size, void* d_ws, size_t ws_size,
                              hipStream_t stream) {
  (void)in_sizes; (void)n_in; (void)out_size; (void)ws_size;

  const float* x_all      = (const float*)d_in[0];   // [9, N, 64]
  const float* ex_all     = (const float*)d_in[1];   // [9, E, 32]
  const int*   hyper_all  = (const int*)  d_in[2];   // [9, 2, NNZ]
  const int*   marks_all  = (const int*)  d_in[3];   // [9, M]
  const int*   emarks_all = (const int*)  d_in[4];   // [9, M]
  const float* Wn  = (const float*)d_in[5];
  const float* bn  = (const float*)d_in[6];
  const float* We  = (const float*)d_in[7];
  const float* be  = (const float*)d_in[8];
  const float* Wcn = (const float*)d_in[9];
  const float* bcn = (const float*)d_in[10];
  const float* Wce = (const float*)d_in[11];
  const float* bce = (const float*)d_in[12];
  const float* pos = (const float*)d_in[13];
  const float* Wq  = (const float*)d_in[14];
  const float* Wk  = (const float*)d_in[15];
  const float* Wv  = (const float*)d_in[16];
  const float* ffW = (const float*)d_in[17];
  const float* ffb = (const float*)d_in[18];
  const float* W1  = (const float*)d_in[19];
  const float* b1  = (const float*)d_in[20];
  const float* W2  = (const float*)d_in[21];
  const float* b2  = (const float*)d_in[22];
  float* out = (float*)d_out;

  // ---- workspace carve-up (floats) ----
  float* wsf = (float*)d_ws;
  size_t off = 0;
  auto alloc = [&](size_t n) { float* p = wsf + off; off += n; return p; };
  float* TI   = alloc((size_t)CM * CT * 512);  // ti / s buffer [M, T, 512]
  float* HIN  = alloc((size_t)CM * 1024);      // [temporal_out | collapsed]
  float* Q7   = alloc((size_t)CM * 512);
  float* QK   = alloc((size_t)CM * 512);
  float* UBUF = alloc((size_t)CM * 512);
  float* O7   = alloc((size_t)CM * 512);
  float* HB   = alloc((size_t)CM * 256);
  float* WKT  = alloc((size_t)512 * 512);
  float* X    = alloc((size_t)CN * 64);
  float* XT   = alloc((size_t)CN * 64);
  float* AGG  = alloc((size_t)CN * 64);
  float* OUTB = alloc((size_t)CN * 64);
  float* EXB  = alloc((size_t)CE * 32);
  float* DEGN = alloc((size_t)CN);
  float* DEGE = alloc((size_t)CE);

  auto zero = [&](float* p, int n) {
    zero_f32<<<(n + 255) / 256, 256, 0, stream>>>(p, n);
  };

  // ================= structural: graphs 0..7 (shared weights) + 8 (collapsed)
  for (int g = 0; g < CT + 1; ++g) {
    const int* ni     = hyper_all + (size_t)g * 2 * CNNZ;  // node indices
    const int* ei     = ni + CNNZ;                         // hyperedge indices
    const int* marks  = marks_all  + (size_t)g * CM;
    const int* emarks = emarks_all + (size_t)g * CM;
    const float* Wn_ = (g < CT) ? Wn : Wcn;
    const float* bn_ = (g < CT) ? bn : bcn;
    const float* We_ = (g < CT) ? We : Wce;
    const float* be_ = (g < CT) ? be : bce;

    hipMemcpyAsync(X,   x_all  + (size_t)g * CN * 64, (size_t)CN * 64 * sizeof(float),
                   hipMemcpyDeviceToDevice, stream);
    hipMemcpyAsync(EXB, ex_all + (size_t)g * CE * 32, (size_t)CE * 32 * sizeof(float),
                   hipMemcpyDeviceToDevice, stream);

    zero(DEGN, CN); zero(DEGE, CE);
    count_deg<<<(CNNZ + 255) / 256, 256, 0, stream>>>(ni, DEGN, CNNZ);
    count_deg<<<(CNNZ + 255) / 256, 256, 0, stream>>>(ei, DEGE, CNNZ);
    invert_f32<<<(CN + 255) / 256, 256, 0, stream>>>(DEGN, CN);
    invert_f32<<<(CE + 255) / 256, 256, 0, stream>>>(DEGE, CE);

    const int nstride = (g < CT) ? (CT * 512) : 1024;
    float* dstBase = (g < CT) ? (TI + (size_t)g * 512) : (HIN + 512);

    for (int lv = 0; lv < CL; ++lv) {
      // ---- node hconv: x = relu(Dinv * H (Binv * H^T (x @ Wn)) + bn)
      launch_gemm(stream, X, Wn_ + (size_t)lv * 64 * 64, XT, CN, 64, 64, 64, 64, 64,
                  nullptr, 0);
      zero(AGG, CN * 64);
      scatter_add<<<(CNNZ * 64 + 255) / 256, 256, 0, stream>>>(XT, AGG, ni, ei,
                                                               nullptr, 6, CNNZ * 64);
      zero(OUTB, CN * 64);
      scatter_add<<<(CNNZ * 64 + 255) / 256, 256, 0, stream>>>(AGG, OUTB, ei, ni,
                                                               DEGE, 6, CNNZ * 64);
      finalize_relu<<<(CN * 64 + 255) / 256, 256, 0, stream>>>(X, OUTB, DEGN,
                                                               bn_ + (size_t)lv * 64,
                                                               6, CN * 64);
      gather_rows<<<(CM * 64 + 255) / 256, 256, 0, stream>>>(
          X, marks, dstBase + 256 + lv * 64, 6, nstride, CM * 64);

      // ---- edge hconv with flipped incidence
      launch_gemm(stream, EXB, We_ + (size_t)lv * 32 * 32, XT, CE, 32, 32, 32, 32, 32,
                  nullptr, 0);
      zero(AGG, CE * 32);
      scatter_add<<<(CNNZ * 32 + 255) / 256, 256, 0, stream>>>(XT, AGG, ei, ni,
                                                               nullptr, 5, CNNZ * 32);
      zero(OUTB, CE * 32);
      scatter_add<<<(CNNZ * 32 + 255) / 256, 256, 0, stream>>>(AGG, OUTB, ni, ei,
                                                               DEGN, 5, CNNZ * 32);
      finalize_relu<<<(CE * 32 + 255) / 256, 256, 0, stream>>>(EXB, OUTB, DEGE,
                                                               be_ + (size_t)lv * 32,
                                                               5, CE * 32);
      gather_edge_mm<<<(CM * 32 + 255) / 256, 256, 0, stream>>>(
          EXB, emarks, dstBase + lv * 32, dstBase + 128 + lv * 32, nstride, CM * 32);
    }
  }

  // ================= temporal attention (only t = T-1 is consumed)
  add_pos<<<(CM * CT * 512 + 255) / 256, 256, 0, stream>>>(TI, pos, CM * CT * 512);
  transpose512<<<(512 * 512 + 255) / 256, 256, 0, stream>>>(Wk, WKT);

  // q7 = ti[:,7,:] @ Wq            [M,512]
  launch_gemm(stream, TI + 7 * 512, Wq, Q7, CM, 512, 512, CT * 512, 512, 512, nullptr, 0);
  // qk = q7 @ Wk^T                 [M,512]  (scores[m,s] = ti[m,s,:].qk[m,:])
  launch_gemm(stream, Q7, WKT, QK, CM, 512, 512, 512, 512, 512, nullptr, 0);
  // softmax row + u = attn-weighted sum of ti rows
  attn_last_row<<<(CM + 7) / 8, 256, 0, stream>>>(TI, QK, UBUF, CM);
  // o7 = u @ Wv                    [M,512]
  launch_gemm(stream, UBUF, Wv, O7, CM, 512, 512, 512, 512, 512, nullptr, 0);
  // HIN[:, 0:512] = o7 @ ffW  (then + ffb + residual ti[:,7,:])
  launch_gemm(stream, O7, ffW, HIN, CM, 512, 512, 512, 512, 1024, nullptr, 0);
  ff_epilogue<<<(CM * 512 + 255) / 256, 256, 0, stream>>>(HIN, TI, ffb, CM * 512);

  // ================= head MLP
  launch_gemm(stream, HIN, W1, HB, CM, 1024, 256, 1024, 256, 256, b1, 1);
  final_logits<<<(CM + 255) / 256, 256, 0, stream>>>(HB, W2, b2, out, CM);
}